// BayesianVNETDetector_61357902791096
// MI455X (gfx1250) — compile-verified
//
#include <hip/hip_runtime.h>

// ---------------------------------------------------------------------------
// BayesianVNET detector for MI455X (gfx1250, wave32).
//   K1: fused MLP+dropout-count -> priors[T,16] via V_WMMA_F32_16X16X4_F32
//   K2: per-chunk 16x16 min-plus aggregates; priors staged via TDM -> LDS
//   K3: sequential fold of 1024 aggregates -> per-chunk start vectors
//   K4: per-chunk replay (det/conf); priors staged via TDM -> LDS
// ---------------------------------------------------------------------------

typedef __attribute__((ext_vector_type(2))) float        v2f;
typedef __attribute__((ext_vector_type(8))) float        v8f;
typedef __attribute__((ext_vector_type(4))) unsigned int v4u;
typedef __attribute__((ext_vector_type(8))) int          v8i;
typedef __attribute__((ext_vector_type(4))) int          v4i;

#define T_LEN   262144
#define HID     75
#define KPAD    76                  // pad K to multiple of 4 for 16x16x4 WMMA
#define NST     16
#define NCHUNK  1024
#define CLEN    (T_LEN / NCHUNK)    // 256 steps per chunk
#define STG     64                  // rows staged per TDM transfer (4 KB)
#define NBLK    (CLEN / STG)        // 4 stages per chunk
#define BIG     3.0e38f

// ---------------- LDS byte offset of a __shared__ pointer ------------------
__device__ __forceinline__ unsigned lds_offset(const void* p) {
    return (unsigned)(unsigned long long)
        (__attribute__((address_space(3))) const void*)p;
}

// ---------------- TDM: 1-D tile of n f32 elements, global -> LDS -----------
// D# per CDNA5 ISA ch.8: group0 {count=1, lds_addr, global_addr[56:0], type=2}
// group1 {data_size=4B, tensor_dim0=n, tensor_dim1=1, tile_dim0=n (1-D)}.
__device__ __forceinline__ void tdm_load_f32(const float* gsrc,
                                             unsigned lds_addr, int n) {
    const unsigned long long ga = (unsigned long long)gsrc;
    v4u g0;
    g0.x = 1u;                                             // count=1 (valid)
    g0.y = lds_addr;                                       // LDS byte address
    g0.z = (unsigned)(ga & 0xFFFFFFFFu);                   // global addr lo
    g0.w = (unsigned)((ga >> 32) & 0x01FFFFFFu) | (2u << 30); // hi | type=2
    const unsigned un = (unsigned)n;
    v8i g1;
    g1[0] = (int)(2u << 16);                               // data_size = 4B
    g1[1] = (int)((un & 0xFFFFu) << 16);                   // tensor_dim0 lo16
    g1[2] = (int)(((un >> 16) & 0xFFFFu) | (1u << 16));    // dim0 hi | dim1=1
    g1[3] = (int)((un & 0xFFFFu) << 16);                   // tile_dim0 = n
    g1[4] = 0;                                             // tile_dim1/2 = 0
    g1[5] = (int)un;                                       // dim0_stride lo
    g1[6] = 0;
    g1[7] = 0;
    const v4i z4 = {0, 0, 0, 0};
    const v8i z8 = {0, 0, 0, 0, 0, 0, 0, 0};
    // amdgpu-toolchain (clang-23) 6-arg form:
    // (g0:uint32x4, g1:int32x8, g2:int32x4, g3:int32x4, int32x8, cpol)
    __builtin_amdgcn_tensor_load_to_lds(g0, g1, z4, z4, z8, 0);
}

// -------------------------- Threefry-2x32 (20 rounds) ----------------------
__device__ __forceinline__ unsigned rotl32(unsigned x, int n) {
    return (x << n) | (x >> (32 - n));
}

__device__ __forceinline__ void threefry2x32(unsigned k0, unsigned k1,
                                             unsigned c0, unsigned c1,
                                             unsigned& o0, unsigned& o1) {
    const unsigned ks0 = k0, ks1 = k1, ks2 = k0 ^ k1 ^ 0x1BD11BDAu;
    unsigned x0 = c0 + ks0, x1 = c1 + ks1;
    const int R[8] = {13, 15, 26, 6, 17, 29, 16, 24};
#pragma unroll
    for (int b = 0; b < 5; ++b) {
#pragma unroll
        for (int r = 0; r < 4; ++r) {
            x0 += x1;
            x1 = rotl32(x1, R[(b & 1) * 4 + r]);
            x1 ^= x0;
        }
        const unsigned inj = (unsigned)(b + 1);
        const unsigned m0 = inj % 3u, m1 = (inj + 1u) % 3u;
        x0 += (m0 == 0u) ? ks0 : ((m0 == 1u) ? ks1 : ks2);
        x1 += ((m1 == 0u) ? ks0 : ((m1 == 1u) ? ks1 : ks2)) + inj;
    }
    o0 = x0; o1 = x1;
}

// ----------------- K1: priors = (relu(rx*W1+b1) .* cnt/5) @ W2^T + b2 ------
// One wave computes one 16x16 output tile with fp32 WMMA, K padded to 76.
__global__ void __launch_bounds__(128)
priors_kernel(const float* __restrict__ rx,  const float* __restrict__ W1,
              const float* __restrict__ b1,  const float* __restrict__ W2,
              const float* __restrict__ b2,  const float* __restrict__ dlogit,
              float* __restrict__ priors, int ntiles) {
    __shared__ float sW1[HID], sB1[HID], sP[HID], sB2[NST];
    __shared__ float sW2T[KPAD * NST];   // [k][s] = W2[s][k], zero-padded

    const int tid = threadIdx.x;
    for (int i = tid; i < HID; i += 128) {
        sW1[i] = W1[i];
        sB1[i] = b1[i];
        const float x = dlogit[i];
        sP[i] = 1.0f / (1.0f + __expf(-x));        // keep-probabilities
    }
    for (int i = tid; i < KPAD * NST; i += 128) {
        const int k = i / NST, s = i % NST;
        sW2T[i] = (k < HID) ? W2[s * HID + k] : 0.0f;
    }
    if (tid < NST) sB2[tid] = b2[tid];
    __syncthreads();

    const int wave = tid >> 5;
    const int lane = tid & 31;
    const int tile = blockIdx.x * 4 + wave;
    if (tile >= ntiles) return;                    // wave-uniform

    const int col  = lane & 15;
    const int m    = tile * 16 + col;              // A-matrix row (time index)
    const int kofs = (lane < 16) ? 0 : 2;          // f32 A/B VGPR K split
    const float rxv = rx[m];

    v8f acc = {};
    for (int kb = 0; kb < KPAD; kb += 4) {
        v2f a, b;
#pragma unroll
        for (int q = 0; q < 2; ++q) {
            const int k = kb + kofs + q;
            float av = 0.0f;
            if (k < HID) {
                float h = rxv * sW1[k] + sB1[k];
                h = (h > 0.0f) ? h : 0.0f;
                unsigned x0, x1;
                threefry2x32(0u, 42u, (unsigned)(m * HID + k), 0u, x0, x1);
                const float pk = sP[k];
                int cnt = 0;
                unsigned s = x0;
#pragma unroll
                for (int e = 0; e < 5; ++e) {       // 5 Bernoulli draws
                    s = s * 1664525u + (x1 ^ ((unsigned)e * 0x9E3779B9u)) + 1013904223u;
                    const float u = (float)s * 2.3283064365386963e-10f;
                    cnt += (u < pk) ? 1 : 0;
                }
                av = h * (float)cnt * 0.2f;         // fold ensemble avg into A
            }
            const float bv = sW2T[k * NST + col];
            if (q == 0) { a.x = av; b.x = bv; } else { a.y = av; b.y = bv; }
        }
        acc = __builtin_amdgcn_wmma_f32_16x16x4_f32(
            false, a, false, b, (short)0, acc, false, false);
    }

    // D layout: VGPR r -> rows r (lanes 0-15) / r+8 (lanes 16-31), N = lane%16
    const int rbase = tile * 16 + ((lane < 16) ? 0 : 8);
#pragma unroll
    for (int r = 0; r < 8; ++r)
        priors[(rbase + r) * NST + col] = acc[r] + sB2[col];
}

// ------------- K2: chunk aggregates (min-plus scan, pass 1) ----------------
// Lane j holds column j of the 16x16 aggregate; upper half mirrors lower.
// Priors are double-buffer staged into LDS by the Tensor Data Mover.
__global__ void __launch_bounds__(256)
agg_kernel(const float* __restrict__ priors, float* __restrict__ agg) {
    __shared__ float stage[8][2][STG * NST];        // 64 KB

    const int w    = threadIdx.x >> 5;
    const int wave = blockIdx.x * 8 + w;
    const int lane = threadIdx.x & 31;
    const int col  = lane & 15;
    if (wave >= NCHUNK) return;

    float A[NST];
#pragma unroll
    for (int k = 0; k < NST; ++k) A[k] = (k == col) ? 0.0f : BIG;

    const float* src = priors + (size_t)wave * CLEN * NST;
    const unsigned lds0 = lds_offset(&stage[w][0][0]);
    const unsigned lds1 = lds_offset(&stage[w][1][0]);

    tdm_load_f32(src, lds0, STG * NST);             // prime stage 0
    for (int blk = 0; blk < NBLK; ++blk) {
        if (blk + 1 < NBLK) {
            tdm_load_f32(src + (size_t)(blk + 1) * STG * NST,
                         ((blk + 1) & 1) ? lds1 : lds0, STG * NST);
            __builtin_amdgcn_s_wait_tensorcnt(1);   // current stage landed
        } else {
            __builtin_amdgcn_s_wait_tensorcnt(0);
        }
        const float* buf = &stage[w][blk & 1][0];
        for (int s = 0; s < STG; ++s) {
            const float4* P4 = (const float4*)(buf + s * NST);
            const float4 q0 = P4[0], q1 = P4[1], q2 = P4[2], q3 = P4[3];
            const float p[NST] = {q0.x, q0.y, q0.z, q0.w, q1.x, q1.y, q1.z, q1.w,
                                  q2.x, q2.y, q2.z, q2.w, q3.x, q3.y, q3.z, q3.w};
            float W[NST];
#pragma unroll
            for (int k = 0; k < NST; ++k) W[k] = A[k] - p[k];
#pragma unroll
            for (int i = 0; i < NST; ++i)
                A[i] = fminf(W[(2 * i) & 15], W[(2 * i + 1) & 15]);
        }
    }
    if (lane < 16) {
#pragma unroll
        for (int k = 0; k < NST; ++k)
            agg[(size_t)wave * 256 + k * NST + col] = A[k];
    }
}

// ------------- K3: fold aggregates -> per-chunk start vectors --------------
// Single wave; software-pipelined: row(c+1) loads overlap matvec(c).
__global__ void __launch_bounds__(32)
scan_kernel(const float* __restrict__ agg, float* __restrict__ start) {
    const int lane = threadIdx.x & 31;
    const int i = lane & 15;
    float v = 0.0f;                                 // in_prob0 = zeros

    const float4* R = (const float4*)(agg + (size_t)i * NST);
    float4 r0 = R[0], r1 = R[1], r2 = R[2], r3 = R[3];

    for (int c = 0; c < NCHUNK; ++c) {
        if (lane < 16) start[c * NST + i] = v;
        const float row[NST] = {r0.x, r0.y, r0.z, r0.w, r1.x, r1.y, r1.z, r1.w,
                                r2.x, r2.y, r2.z, r2.w, r3.x, r3.y, r3.z, r3.w};
        if (c + 1 < NCHUNK) {                       // issue next row's loads
            const float4* Rn =
                (const float4*)(agg + (size_t)(c + 1) * 256 + i * NST);
            r0 = Rn[0]; r1 = Rn[1]; r2 = Rn[2]; r3 = Rn[3];
        }
        float nv = BIG;
#pragma unroll
        for (int j = 0; j < NST; ++j) {             // overlaps load latency
            const float vj = __shfl(v, j, 32);
            nv = fminf(nv, row[j] + vj);
        }
        v = nv;
    }
}

// ------------- K4: replay each chunk, emit det + confidence ----------------
// Priors staged into LDS by TDM (double-buffered) to shorten the per-step
// dependency chain of the sequential recurrence.
__global__ void __launch_bounds__(256)
replay_kernel(const float* __restrict__ priors, const float* __restrict__ start,
              float* __restrict__ det_out, float* __restrict__ conf_out) {
    __shared__ float stage[8][2][STG * NST];        // 64 KB

    const int w    = threadIdx.x >> 5;
    const int wave = blockIdx.x * 8 + w;
    const int lane = threadIdx.x & 31;
    const int j    = lane & 15;                     // state index (mirrored)
    if (wave >= NCHUNK) return;

    float v = start[wave * NST + j];
    const int base = wave * CLEN;
    const int half = lane & 16;

    const float* src = priors + (size_t)wave * CLEN * NST;
    const unsigned lds0 = lds_offset(&stage[w][0][0]);
    const unsigned lds1 = lds_offset(&stage[w][1][0]);

    tdm_load_f32(src, lds0, STG * NST);             // prime stage 0
    for (int blk = 0; blk < NBLK; ++blk) {
        if (blk + 1 < NBLK) {
            tdm_load_f32(src + (size_t)(blk + 1) * STG * NST,
                         ((blk + 1) & 1) ? lds1 : lds0, STG * NST);
            __builtin_amdgcn_s_wait_tensorcnt(1);
        } else {
            __builtin_amdgcn_s_wait_tensorcnt(0);
        }
        const float* buf = &stage[w][blk & 1][0];

        for (int s = 0; s < STG; ++s) {
            const int t = base + blk * STG + s;

            // argmin with first-index tie-break; 4 xor rounds per 16-half
            float mv = v; int mi = j;
#pragma unroll
            for (int d = 1; d < 16; d <<= 1) {
                const float ov = __shfl_xor(mv, d, 32);
                const int   oi = __shfl_xor(mi, d, 32);
                if (ov < mv || (ov == mv && oi < mi)) { mv = ov; mi = oi; }
            }
            // conf = max softmax(-v) = 1 / sum_j exp(min(v) - v_j)
            float e = __expf(mv - v);
#pragma unroll
            for (int d = 1; d < 16; d <<= 1) e += __shfl_xor(e, d, 32);
            if (lane == 0) {
                det_out[t]  = (float)(mi & 1);
                conf_out[t] = 1.0f / e;
            }
            // ACS step: w = v - p;  v'[i] = min(w[2i%16], w[(2i+1)%16])
            const float p = buf[s * NST + j];
            const float wv = v - p;
            const float w0 = __shfl(wv, ((2 * j) & 15) | half, 32);
            const float w1 = __shfl(wv, ((2 * j + 1) & 15) | half, 32);
            v = fminf(w0, w1);
        }
    }
}

// ---------------------------------------------------------------------------
extern "C" void kernel_launch(void* const* d_in, const int* in_sizes, int n_in,
                              void* d_out, int out_size, void* d_ws, size_t ws_size,
                              hipStream_t stream) {
    const float* rx    = (const float*)d_in[0];
    const float* W1    = (const float*)d_in[1];
    const float* b1    = (const float*)d_in[2];
    const float* W2    = (const float*)d_in[3];
    const float* b2    = (const float*)d_in[4];
    const float* dlog  = (const float*)d_in[5];
    float* out = (float*)d_out;

    char* ws = (char*)d_ws;
    float* priors = (float*)ws;                                     // 16 MB
    float* agg    = (float*)(ws + (size_t)T_LEN * NST * 4);         //  1 MB
    float* start  = (float*)(ws + (size_t)T_LEN * NST * 4
                                + (size_t)NCHUNK * 256 * 4);        // 64 KB

    priors_kernel<<<(T_LEN / 16) / 4, 128, 0, stream>>>(
        rx, W1, b1, W2, b2, dlog, priors, T_LEN / 16);
    agg_kernel<<<NCHUNK / 8, 256, 0, stream>>>(priors, agg);
    scan_kernel<<<1, 32, 0, stream>>>(agg, start);
    replay_kernel<<<NCHUNK / 8, 256, 0, stream>>>(priors, start,
                                                  out, out + T_LEN);
}